// ACN2dMultiBranch_67946382623352
// MI455X (gfx1250) — compile-verified
//
#include <hip/hip_runtime.h>

typedef float v2f __attribute__((ext_vector_type(2)));
typedef float v8f __attribute__((ext_vector_type(8)));

// Problem sizes fixed by setup_inputs()
constexpr int B = 8, C = 128, N = 32768, G = 8;
constexpr float EPS = 1e-3f;       // inside sqrt
constexpr float A_NORM_EPS = 1e-8f;
constexpr int NSPLIT = 16;         // N partitions in pass 2

// ---------------------------------------------------------------------------
// Pass 1: att[b,g,n] = sum_c W[g,c] x[b,c,n] + bias[g]; softmax over g;
// store a[b,g,n]. Streaming, memory-bound; W (4 KB) staged in LDS (broadcast).
// ---------------------------------------------------------------------------
__global__ __launch_bounds__(256) void k_pass1(const float* __restrict__ x,
                                               const float* __restrict__ w,
                                               const float* __restrict__ bias,
                                               float* __restrict__ a) {
  __shared__ float wS[G * C];
  int tid = threadIdx.x;
  for (int i = tid; i < G * C; i += 256) wS[i] = w[i];
  __syncthreads();

  int b = blockIdx.y;
  int n = blockIdx.x * 256 + tid;

  float att[G];
#pragma unroll
  for (int g = 0; g < G; ++g) att[g] = bias[g];

  const float* xb = x + (size_t)b * C * N + n;
#pragma unroll 4
  for (int c = 0; c < C; ++c) {
    float xv = xb[(size_t)c * N];
#pragma unroll
    for (int g = 0; g < G; ++g) att[g] = fmaf(wS[g * C + c], xv, att[g]);
  }

  float mx = att[0];
#pragma unroll
  for (int g = 1; g < G; ++g) mx = fmaxf(mx, att[g]);
  float e[G];
  float sum = 0.f;
#pragma unroll
  for (int g = 0; g < G; ++g) { e[g] = __expf(att[g] - mx); sum += e[g]; }
  float inv = 1.f / sum;

  float* ab = a + (size_t)b * G * N + n;
#pragma unroll
  for (int g = 0; g < G; ++g) ab[(size_t)g * N] = e[g] * inv;
}

// ---------------------------------------------------------------------------
// Sum of a over n per (b,g): s[b*G+g]. One block per (b,g); deterministic.
// ---------------------------------------------------------------------------
__global__ __launch_bounds__(256) void k_sums(const float* __restrict__ a,
                                              float* __restrict__ s) {
  __shared__ float red[256];
  int bg = blockIdx.x;
  const float* ap = a + (size_t)bg * N;
  float v = 0.f;
  for (int n = threadIdx.x; n < N; n += 256) v += ap[n];
  red[threadIdx.x] = v;
  __syncthreads();
  for (int st = 128; st > 0; st >>= 1) {
    if (threadIdx.x < st) red[threadIdx.x] += red[threadIdx.x + st];
    __syncthreads();
  }
  if (threadIdx.x == 0) s[bg] = red[0];
}

// ---------------------------------------------------------------------------
// Pass 2: Sx[b,c,g] = sum_n x*a, Sxx[b,c,g] = sum_n x^2*a via
// V_WMMA_F32_16X16X4_F32.  M = 16 channels, Ncols = 16 groups (g>=8 masked 0),
// K = n in steps of 4.  Grid (NSPLIT, C/16, B), 8 waves split the n range.
// Per-block partials written to Px/Pxx (no float atomics -> deterministic).
// A-frag f32 16x4: lane l -> row c=l%16, V0=K(2*(l/16)), V1=K(+1) (b64 load).
// B-frag 4x16 mirrors this (col g=l%16).
// ---------------------------------------------------------------------------
__global__ __launch_bounds__(256) void k_pass2(const float* __restrict__ x,
                                               const float* __restrict__ a,
                                               float* __restrict__ Px,
                                               float* __restrict__ Pxx) {
  constexpr int NPB = N / NSPLIT;      // 2048 n per block
  constexpr int NPW = NPB / 8;         // 256 n per wave
  __shared__ float red[2][8][256];

  int b = blockIdx.z, ct = blockIdx.y;
  int tid = threadIdx.x;
  int wave = tid >> 5, lane = tid & 31;
  int row = lane & 15, half = lane >> 4, koff = half * 2;

  const float* xrow = x + ((size_t)b * C + ct * 16 + row) * N;       // A rows
  const float* arow = a + ((size_t)b * G + (row & 7)) * N;           // B cols
  float gmask = (row < 8) ? 1.f : 0.f;

  int n0 = blockIdx.x * NPB + wave * NPW;

  v8f acc1 = {};
  v8f acc2 = {};
  for (int k = 0; k < NPW / 4; ++k) {
    int n = n0 + 4 * k + koff;                    // 8-byte aligned
    v2f av = *(const v2f*)(xrow + n);
    v2f bv = *(const v2f*)(arow + n);
    bv.x *= gmask; bv.y *= gmask;
    v2f sq; sq.x = av.x * av.x; sq.y = av.y * av.y;
    acc1 = __builtin_amdgcn_wmma_f32_16x16x4_f32(false, av, false, bv,
                                                 (short)0, acc1, false, false);
    acc2 = __builtin_amdgcn_wmma_f32_16x16x4_f32(false, sq, false, bv,
                                                 (short)0, acc2, false, false);
  }

#pragma unroll
  for (int v = 0; v < 8; ++v) {
    red[0][wave][v * 32 + lane] = acc1[v];
    red[1][wave][v * 32 + lane] = acc2[v];
  }
  __syncthreads();

  // reduce the 8 wave partials; thread t owns (vgpr=t/32, lane=t%32)
  float s1 = 0.f, s2 = 0.f;
#pragma unroll
  for (int wv = 0; wv < 8; ++wv) { s1 += red[0][wv][tid]; s2 += red[1][wv][tid]; }
  int v = tid >> 5, l = tid & 31, rr = l & 15, h = l >> 4;
  if (rr < 8) {                                   // only real groups
    size_t o = ((size_t)b * C + ct * 16 + v + 8 * h) * G + rr;
    size_t po = (size_t)blockIdx.x * (B * C * G) + o;
    Px[po] = s1;
    Pxx[po] = s2;
  }
}

// ---------------------------------------------------------------------------
// Stats: reduce NSPLIT partials, normalize by (s+1e-8), variance identity,
// inv_std = rsqrt(var + 1e-3). Outputs istd and mean*istd.
// ---------------------------------------------------------------------------
__global__ __launch_bounds__(256) void k_stats(const float* __restrict__ Px,
                                               const float* __restrict__ Pxx,
                                               const float* __restrict__ s,
                                               float* __restrict__ istd,
                                               float* __restrict__ minv) {
  int i = blockIdx.x * 256 + threadIdx.x;
  if (i >= B * C * G) return;
  int g = i % G;
  int b = i / (C * G);
  float sx = 0.f, sxx = 0.f;
#pragma unroll
  for (int p = 0; p < NSPLIT; ++p) {
    sx += Px[(size_t)p * (B * C * G) + i];
    sxx += Pxx[(size_t)p * (B * C * G) + i];
  }
  float sv = s[b * G + g];
  float denom = sv + A_NORM_EPS;
  float T = sv / denom;
  float mean = sx / denom;
  float m2 = sxx / denom;
  float var = m2 - 2.f * mean * mean + mean * mean * T;
  float is = rsqrtf(var + EPS);
  istd[i] = is;
  minv[i] = mean * is;
}

// ---------------------------------------------------------------------------
// Pass 3: A1[c,n] = sum_g istd[c,g] a[g,n];  A2 likewise with mean*istd;
// out = x*A1 - A2.  K = G = 8 -> exactly two 16x16x4 WMMA steps, epilogue
// fused in the C/D layout (row c = v + 8*(lane/16), col n = lane%16).
// ---------------------------------------------------------------------------
__global__ __launch_bounds__(256) void k_pass3(const float* __restrict__ x,
                                               const float* __restrict__ a,
                                               const float* __restrict__ istd,
                                               const float* __restrict__ minv,
                                               float* __restrict__ out) {
  int b = blockIdx.z, ct = blockIdx.y;
  int wave = threadIdx.x >> 5, lane = threadIdx.x & 31;
  int ntile = blockIdx.x * 8 + wave;
  int row = lane & 15, half = lane >> 4, koff = half * 2;

  const float* iptr = istd + ((size_t)b * C + ct * 16 + row) * G;
  const float* mptr = minv + ((size_t)b * C + ct * 16 + row) * G;
  int n = ntile * 16 + row;
  const float* ab = a + (size_t)b * G * N + n;

  v8f A1 = {};
  v8f A2 = {};
#pragma unroll
  for (int k = 0; k < 2; ++k) {
    int ks = 4 * k + koff;
    v2f fi = *(const v2f*)(iptr + ks);            // A-frag (istd)
    v2f fm = *(const v2f*)(mptr + ks);            // A-frag (mean*istd)
    v2f fb; fb.x = ab[(size_t)ks * N]; fb.y = ab[(size_t)(ks + 1) * N]; // B-frag
    A1 = __builtin_amdgcn_wmma_f32_16x16x4_f32(false, fi, false, fb,
                                               (short)0, A1, false, false);
    A2 = __builtin_amdgcn_wmma_f32_16x16x4_f32(false, fm, false, fb,
                                               (short)0, A2, false, false);
  }

  size_t base = ((size_t)b * C + ct * 16 + 8 * half) * N + (size_t)ntile * 16 + row;
#pragma unroll
  for (int v = 0; v < 8; ++v) {
    float xv = x[base + (size_t)v * N];
    out[base + (size_t)v * N] = xv * A1[v] - A2[v];
  }
}

// ---------------------------------------------------------------------------
extern "C" void kernel_launch(void* const* d_in, const int* in_sizes, int n_in,
                              void* d_out, int out_size, void* d_ws, size_t ws_size,
                              hipStream_t stream) {
  (void)in_sizes; (void)n_in; (void)out_size; (void)ws_size;
  const float* x = (const float*)d_in[0];     // [B,C,N,1]
  const float* w = (const float*)d_in[1];     // [G,C]
  const float* bias = (const float*)d_in[2];  // [G]
  float* out = (float*)d_out;                 // [B,C,N,1]

  char* p = (char*)d_ws;
  float* a = (float*)p;     p += (size_t)B * G * N * sizeof(float);       // 8 MB
  float* s = (float*)p;     p += 256;                                     // B*G
  float* Px = (float*)p;    p += (size_t)NSPLIT * B * C * G * sizeof(float);
  float* Pxx = (float*)p;   p += (size_t)NSPLIT * B * C * G * sizeof(float);
  float* istd = (float*)p;  p += (size_t)B * C * G * sizeof(float);
  float* minv = (float*)p;

  k_pass1<<<dim3(N / 256, B), 256, 0, stream>>>(x, w, bias, a);
  k_sums<<<B * G, 256, 0, stream>>>(a, s);
  k_pass2<<<dim3(NSPLIT, C / 16, B), 256, 0, stream>>>(x, a, Px, Pxx);
  k_stats<<<(B * C * G + 255) / 256, 256, 0, stream>>>(Px, Pxx, s, istd, minv);
  k_pass3<<<dim3(N / 128, C / 16, B), 256, 0, stream>>>(x, a, istd, minv, out);
}